// GAT_h_20323785244856
// MI455X (gfx1250) — compile-verified
//
#include <hip/hip_runtime.h>
#include <math.h>

typedef __attribute__((ext_vector_type(2))) float v2f;
typedef __attribute__((ext_vector_type(8))) float v8f;

#define H8 8
#define NEG_SLOPE 0.2f

// ---------------------------------------------------------------------------
// order-preserving float<->uint encoding for atomicMax-based segment max
// ---------------------------------------------------------------------------
__device__ __forceinline__ unsigned fenc(float f) {
    unsigned u = __float_as_uint(f);
    return (u & 0x80000000u) ? ~u : (u | 0x80000000u);
}
__device__ __forceinline__ float fdec(unsigned k) {
    unsigned u = (k & 0x80000000u) ? (k & 0x7FFFFFFFu) : ~k;
    return __uint_as_float(u);
}
__device__ __forceinline__ float lrelu(float v) { return v > 0.0f ? v : NEG_SLOPE * v; }

// ---------------------------------------------------------------------------
// WMMA fp32 GEMM: C[M,128] = epilogue(A'[M,128] @ B[128,128])
//   A'[m,k] = PRESCALE ? A[m,k]*scale[k]+shift[k] : A[m,k]
//   epilogue: +bias[n] (if bias), ReLU (if RELU)
// Block = 256 threads = 8 waves, covers 32 rows x 64 cols (2x4 wave tiles).
// grid = (ceil(M/32), 2). B strip + A strip staged in LDS (51.7 KB).
// ---------------------------------------------------------------------------
#define LDA_S 132   // padded LDS stride for A (bank-conflict-free frag reads)
#define LDB_S 68    // padded LDS stride for B

template <bool RELU, bool PRESCALE>
__global__ __launch_bounds__(256) void gemm128_wmma(
    const float* __restrict__ A, const float* __restrict__ B,
    const float* __restrict__ bias,
    const float* __restrict__ scale, const float* __restrict__ shift,
    float* __restrict__ C, int M)
{
    __shared__ float sA[32 * LDA_S];
    __shared__ float sB[128 * LDB_S];

    const int t  = threadIdx.x;
    const int m0 = blockIdx.x * 32;
    const int nb = blockIdx.y * 64;

    // cooperative load of A strip (32 x 128), optionally pre-scaled per column k
    {
        const int row0 = t >> 5;          // 0..7
        const int col  = (t & 31) * 4;    // 0..124 (k index)
        float4 sc, sh;
        if (PRESCALE) {
            sc = *(const float4*)(scale + col);
            sh = *(const float4*)(shift + col);
        }
        #pragma unroll
        for (int it = 0; it < 4; ++it) {
            const int r  = row0 + it * 8;
            int gm = m0 + r; if (gm >= M) gm = M - 1;   // clamp (keeps EXEC full)
            float4 v = *(const float4*)(A + (size_t)gm * 128 + col);
            if (PRESCALE) {
                v.x = v.x * sc.x + sh.x;  v.y = v.y * sc.y + sh.y;
                v.z = v.z * sc.z + sh.z;  v.w = v.w * sc.w + sh.w;
            }
            *(float4*)(&sA[r * LDA_S + col]) = v;
        }
    }
    // cooperative load of B strip (128 x 64)
    {
        const int row0 = t >> 4;          // 0..15
        const int col  = (t & 15) * 4;    // 0..60
        #pragma unroll
        for (int it = 0; it < 8; ++it) {
            const int r = row0 + it * 16;
            *(float4*)(&sB[r * LDB_S + col]) =
                *(const float4*)(B + (size_t)r * 128 + nb + col);
        }
    }
    __syncthreads();

    const int wave  = t >> 5;             // 0..7
    const int lane  = t & 31;
    const int lhalf = lane >> 4;          // 0 | 1
    const int lmod  = lane & 15;
    const int mt    = (wave >> 2) * 16;   // row tile within strip
    const int nt    = (wave & 3) * 16;    // col tile within strip

    v8f acc = {};
    #pragma unroll
    for (int k0 = 0; k0 < 128; k0 += 4) {
        const int ka = k0 + lhalf * 2;    // lanes 0-15: k0,k0+1 ; lanes 16-31: k0+2,k0+3
        v2f a = *(const v2f*)(&sA[(mt + lmod) * LDA_S + ka]);
        v2f b;
        b.x = sB[(ka    ) * LDB_S + nt + lmod];
        b.y = sB[(ka + 1) * LDB_S + nt + lmod];
        acc = __builtin_amdgcn_wmma_f32_16x16x4_f32(
            /*neg_a=*/false, a, /*neg_b=*/false, b,
            /*c_mod=*/(short)0, acc, /*reuse_a=*/false, /*reuse_b=*/false);
    }

    // D layout: vgpr r -> row r (lanes 0-15) / row 8+r (lanes 16-31), col = lmod
    const int gn = nb + nt + lmod;
    #pragma unroll
    for (int r = 0; r < 8; ++r) {
        const int gm = m0 + mt + lhalf * 8 + r;
        if (gm < M) {
            float v = acc[r];
            if (bias) v += bias[gn];
            if (RELU) v = fmaxf(v, 0.0f);
            C[(size_t)gm * 128 + gn] = v;
        }
    }
}

// ---------------------------------------------------------------------------
// a[n,h] = sum_c x[n, h*16+c] * att[h*16+c]
// ---------------------------------------------------------------------------
__global__ void att_scores(const float* __restrict__ x, const float* __restrict__ att,
                           float* __restrict__ out, int N)
{
    int gid = blockIdx.x * blockDim.x + threadIdx.x;
    if (gid >= N * H8) return;
    const int n = gid >> 3, h = gid & 7;
    const float4* xr = (const float4*)(x + (size_t)n * 128 + h * 16);
    const float4* ar = (const float4*)(att + h * 16);
    float s = 0.0f;
    #pragma unroll
    for (int i = 0; i < 4; ++i) {
        float4 a = xr[i], b = ar[i];
        s += a.x * b.x + a.y * b.y + a.z * b.z + a.w * b.w;
    }
    out[gid] = s;
}

// U[k,h] = sum_c Wd[k, h*16+c] * attd[h*16+c]   (128 x 8)
__global__ void make_U(const float* __restrict__ Wd, const float* __restrict__ attd,
                       float* __restrict__ U)
{
    int gid = blockIdx.x * blockDim.x + threadIdx.x;
    if (gid >= 128 * H8) return;
    const int k = gid >> 3, h = gid & 7;
    float s = 0.0f;
    #pragma unroll
    for (int c = 0; c < 16; ++c) s += Wd[(size_t)k * 128 + h * 16 + c] * attd[h * 16 + c];
    U[k * H8 + h] = s;
}

// a_d[n,h] = sum_k emb[n,k] * U[k,h]
__global__ void gemv8(const float* __restrict__ emb, const float* __restrict__ U,
                      float* __restrict__ out, int N)
{
    int gid = blockIdx.x * blockDim.x + threadIdx.x;
    if (gid >= N * H8) return;
    const int n = gid >> 3, h = gid & 7;
    const float4* er = (const float4*)(emb + (size_t)n * 128);
    float s = 0.0f;
    #pragma unroll 4
    for (int k4 = 0; k4 < 32; ++k4) {
        float4 e = er[k4];
        const int k = k4 * 4;
        s += e.x * U[(k + 0) * H8 + h] + e.y * U[(k + 1) * H8 + h] +
             e.z * U[(k + 2) * H8 + h] + e.w * U[(k + 3) * H8 + h];
    }
    out[gid] = s;
}

// ---------------------------------------------------------------------------
// init kernels
// ---------------------------------------------------------------------------
__global__ void fill_u32(unsigned* __restrict__ p, unsigned v, long long n) {
    long long i = (long long)blockIdx.x * blockDim.x + threadIdx.x;
    if (i < n) p[i] = v;
}
__global__ void fill_f32(float* __restrict__ p, float v, long long n) {
    long long i = (long long)blockIdx.x * blockDim.x + threadIdx.x;
    if (i < n) p[i] = v;
}
// acc[n, c] = bias1[c] + bias2[c]
__global__ void init_acc(float* __restrict__ acc, const float* __restrict__ b1,
                         const float* __restrict__ b2, int N) {
    long long i = (long long)blockIdx.x * blockDim.x + threadIdx.x;
    if (i < (long long)N * 128) acc[i] = b1[i & 127] + b2[i & 127];
}

// ---------------------------------------------------------------------------
// edge kernels: e in [0,E) from edge list (src=ei[e], dst=ei[E+e]); e>=E -> self loop
// ---------------------------------------------------------------------------
__global__ void edge_max(const int* __restrict__ ei, int E,
                         const float* __restrict__ a_s, const float* __restrict__ a_d,
                         unsigned* __restrict__ menc, long long total)
{
    long long gid = (long long)blockIdx.x * blockDim.x + threadIdx.x;
    if (gid >= total) return;
    const int e = (int)(gid >> 3), h = (int)(gid & 7);
    int s, d;
    if (e < E) { s = ei[e]; d = ei[E + e]; } else { s = d = e - E; }
    const float v = lrelu(a_s[(size_t)s * H8 + h] + a_d[(size_t)d * H8 + h]);
    atomicMax(&menc[(size_t)d * H8 + h], fenc(v));
}

__global__ void edge_exp(const int* __restrict__ ei, int E,
                         const float* __restrict__ a_s, const float* __restrict__ a_d,
                         const unsigned* __restrict__ menc,
                         float* __restrict__ p, float* __restrict__ denom, long long total)
{
    long long gid = (long long)blockIdx.x * blockDim.x + threadIdx.x;
    if (gid >= total) return;
    const int e = (int)(gid >> 3), h = (int)(gid & 7);
    int s, d;
    if (e < E) { s = ei[e]; d = ei[E + e]; } else { s = d = e - E; }
    float m = fdec(menc[(size_t)d * H8 + h]);
    if (!isfinite(m)) m = 0.0f;                        // reference guard
    const float v  = lrelu(a_s[(size_t)s * H8 + h] + a_d[(size_t)d * H8 + h]);
    const float pe = expf(v - m);
    p[gid] = pe;
    atomicAdd(&denom[(size_t)d * H8 + h], pe);
}

// 128 threads per edge (one per feature); block of 256 handles 2 edges
__global__ __launch_bounds__(256) void edge_agg(
    const int* __restrict__ ei, int E,
    const float* __restrict__ xs, const float* __restrict__ p,
    const float* __restrict__ denom, float* __restrict__ acc, long long total)
{
    const int t = threadIdx.x;
    const long long eidx = (long long)blockIdx.x * 2 + (t >> 7);
    if (eidx >= total) return;
    const int e = (int)eidx;
    const int c = t & 127;
    const int h = c >> 4;
    int s, d;
    if (e < E) { s = ei[e]; d = ei[E + e]; } else { s = d = e - E; }
    const float alpha = p[(size_t)e * H8 + h] / (denom[(size_t)d * H8 + h] + 1e-16f);
    atomicAdd(&acc[(size_t)d * 128 + c], xs[(size_t)s * 128 + c] * alpha);
}

// ---------------------------------------------------------------------------
// GraphNorm: one-pass column sums (var via E[x^2] - 2a*E[x] + a^2, a = ms*mu)
// ---------------------------------------------------------------------------
__global__ __launch_bounds__(128) void colstats(const float* __restrict__ x,
                                                float* __restrict__ colsum,
                                                float* __restrict__ colsq, int N)
{
    const int c = threadIdx.x;
    float s = 0.0f, q = 0.0f;
    for (int n = blockIdx.x; n < N; n += gridDim.x) {
        const float v = x[(size_t)n * 128 + c];
        s += v; q += v * v;
    }
    atomicAdd(&colsum[c], s);
    atomicAdd(&colsq[c], q);
}

__global__ void finalize_norm(const float* __restrict__ colsum, const float* __restrict__ colsq,
                              const float* __restrict__ w, const float* __restrict__ b,
                              const float* __restrict__ ms,
                              float* __restrict__ scale, float* __restrict__ shift, int N)
{
    const int c = threadIdx.x;
    const float inv = 1.0f / (float)N;
    const float mu  = colsum[c] * inv;
    const float a   = ms[c] * mu;
    const float var = colsq[c] * inv - 2.0f * a * mu + a * a;
    const float rs  = rsqrtf(var + 1e-5f);
    scale[c] = w[c] * rs;
    shift[c] = b[c] - w[c] * a * rs;
}

// out_a rows are identical: v = gnb_author @ W2_author + b2_author
__global__ void author_vec(const float* __restrict__ gnb, const float* __restrict__ W2,
                           const float* __restrict__ b2, float* __restrict__ v)
{
    const int j = threadIdx.x;
    float s = b2[j];
    for (int k = 0; k < 128; ++k) s += gnb[k] * W2[(size_t)k * 128 + j];
    v[j] = s;
}

__global__ void fill_rows(const float* __restrict__ v, float* __restrict__ out, long long n)
{
    long long gid = (long long)blockIdx.x * blockDim.x + threadIdx.x;
    if (gid < n) out[gid] = v[gid & 127];
}

// ---------------------------------------------------------------------------
// host-side orchestration
// ---------------------------------------------------------------------------
static inline int cdiv(long long a, long long b) { return (int)((a + b - 1) / b); }

extern "C" void kernel_launch(void* const* d_in, const int* in_sizes, int n_in,
                              void* d_out, int out_size, void* d_ws, size_t ws_size,
                              hipStream_t stream)
{
    (void)n_in; (void)out_size; (void)ws_size;
    const float* x_paper   = (const float*)d_in[0];
    const float* x_author  = (const float*)d_in[1];
    const int*   ei_w      = (const int*)d_in[2];
    const int*   ei_c      = (const int*)d_in[3];
    const float* W_lin_p   = (const float*)d_in[4];
    const float* b_lin_p   = (const float*)d_in[5];
    const float* W_lin_a   = (const float*)d_in[6];
    const float* b_lin_a   = (const float*)d_in[7];
    const float* Wsrc_w    = (const float*)d_in[8];
    const float* Wdst_w    = (const float*)d_in[9];
    const float* atts_w    = (const float*)d_in[10];
    const float* attd_w    = (const float*)d_in[11];
    const float* bias_w    = (const float*)d_in[12];
    const float* Wsrc_c    = (const float*)d_in[13];
    const float* Wdst_c    = (const float*)d_in[14];
    const float* atts_c    = (const float*)d_in[15];
    const float* attd_c    = (const float*)d_in[16];
    const float* bias_c    = (const float*)d_in[17];
    const float* gnw_p     = (const float*)d_in[18];
    const float* gnb_p     = (const float*)d_in[19];
    const float* gnm_p     = (const float*)d_in[20];
    const float* gnb_a     = (const float*)d_in[22];
    const float* W2_p      = (const float*)d_in[24];
    const float* b2_p      = (const float*)d_in[25];
    const float* W2_a      = (const float*)d_in[26];
    const float* b2_a      = (const float*)d_in[27];

    const int Np = in_sizes[0] / 128;
    const int Na = in_sizes[1] / 128;
    const int Ew = in_sizes[2] / 2;
    const int Ec = in_sizes[3] / 2;
    const int loops_w = (Na < Np) ? Na : Np;
    const int loops_c = Np;

    // bump allocator over workspace
    float* ws = (float*)d_ws;
    size_t off = 0;
    auto alloc = [&](size_t n) { float* p = ws + off; off += n; return p; };
    float*    emb_p  = alloc((size_t)Np * 128);
    float*    emb_a  = alloc((size_t)Na * 128);
    float*    xs_w   = alloc((size_t)Na * 128);
    float*    xs_c   = alloc((size_t)Np * 128);
    float*    acc    = alloc((size_t)Np * 128);
    float*    asw    = alloc((size_t)Na * H8);
    float*    adw    = alloc((size_t)Np * H8);
    float*    asc    = alloc((size_t)Np * H8);
    float*    adc    = alloc((size_t)Np * H8);
    unsigned* menc   = (unsigned*)alloc((size_t)Np * H8);
    float*    denom  = alloc((size_t)Np * H8);
    const size_t pcap = ((size_t)((Ew > Ec) ? Ew : Ec) + (size_t)Np) * H8;
    float*    pbuf   = alloc(pcap);
    float*    U      = alloc(128 * H8);
    float*    colsum = alloc(128);
    float*    colsq  = alloc(128);
    float*    scalev = alloc(128);
    float*    shiftv = alloc(128);
    float*    avec   = alloc(128);

    float* out_p = (float*)d_out;
    float* out_a = out_p + (size_t)Np * 128;

    const dim3 gP(cdiv(Np, 32), 2), gA(cdiv(Na, 32), 2);

    // 1) input projections (fused bias + ReLU)
    gemm128_wmma<true, false><<<gP, 256, 0, stream>>>(x_paper, W_lin_p, b_lin_p, nullptr, nullptr, emb_p, Np);
    gemm128_wmma<true, false><<<gA, 256, 0, stream>>>(x_author, W_lin_a, b_lin_a, nullptr, nullptr, emb_a, Na);

    // 2) source transforms (kept: needed for messages)
    gemm128_wmma<false, false><<<gA, 256, 0, stream>>>(emb_a, Wsrc_w, nullptr, nullptr, nullptr, xs_w, Na);
    gemm128_wmma<false, false><<<gP, 256, 0, stream>>>(emb_p, Wsrc_c, nullptr, nullptr, nullptr, xs_c, Np);

    // 3) attention scores
    att_scores<<<cdiv((long long)Na * H8, 256), 256, 0, stream>>>(xs_w, atts_w, asw, Na);
    att_scores<<<cdiv((long long)Np * H8, 256), 256, 0, stream>>>(xs_c, atts_c, asc, Np);
    make_U<<<4, 256, 0, stream>>>(Wdst_w, attd_w, U);
    gemv8<<<cdiv((long long)Np * H8, 256), 256, 0, stream>>>(emb_p, U, adw, Np);
    make_U<<<4, 256, 0, stream>>>(Wdst_c, attd_c, U);
    gemv8<<<cdiv((long long)Np * H8, 256), 256, 0, stream>>>(emb_p, U, adc, Np);

    // 4) accumulator starts at bias_w + bias_c
    init_acc<<<cdiv((long long)Np * 128, 256), 256, 0, stream>>>(acc, bias_w, bias_c, Np);

    const long long nseg = (long long)Np * H8;
    const unsigned NEG_INF_ENC = 0x007FFFFFu;   // fenc(-inf)

    // 5) writes GAT (author -> paper)
    {
        const long long tot = ((long long)Ew + loops_w) * H8;
        fill_u32<<<cdiv(nseg, 256), 256, 0, stream>>>(menc, NEG_INF_ENC, nseg);
        fill_f32<<<cdiv(nseg, 256), 256, 0, stream>>>(denom, 0.0f, nseg);
        edge_max<<<cdiv(tot, 256), 256, 0, stream>>>(ei_w, Ew, asw, adw, menc, tot);
        edge_exp<<<cdiv(tot, 256), 256, 0, stream>>>(ei_w, Ew, asw, adw, menc, pbuf, denom, tot);
        edge_agg<<<cdiv(Ew + loops_w, 2), 256, 0, stream>>>(ei_w, Ew, xs_w, pbuf, denom, acc, (long long)Ew + loops_w);
    }
    // 6) cites GAT (paper -> paper)
    {
        const long long tot = ((long long)Ec + loops_c) * H8;
        fill_u32<<<cdiv(nseg, 256), 256, 0, stream>>>(menc, NEG_INF_ENC, nseg);
        fill_f32<<<cdiv(nseg, 256), 256, 0, stream>>>(denom, 0.0f, nseg);
        edge_max<<<cdiv(tot, 256), 256, 0, stream>>>(ei_c, Ec, asc, adc, menc, tot);
        edge_exp<<<cdiv(tot, 256), 256, 0, stream>>>(ei_c, Ec, asc, adc, menc, pbuf, denom, tot);
        edge_agg<<<cdiv(Ec + loops_c, 2), 256, 0, stream>>>(ei_c, Ec, xs_c, pbuf, denom, acc, (long long)Ec + loops_c);
    }

    // 7) GraphNorm stats, folded into final GEMM as per-column affine on A
    fill_f32<<<1, 256, 0, stream>>>(colsum, 0.0f, 256);   // colsum+colsq are contiguous
    colstats<<<512, 128, 0, stream>>>(acc, colsum, colsq, Np);
    finalize_norm<<<1, 128, 0, stream>>>(colsum, colsq, gnw_p, gnb_p, gnm_p, scalev, shiftv, Np);
    gemm128_wmma<false, true><<<gP, 256, 0, stream>>>(acc, W2_p, b2_p, scalev, shiftv, out_p, Np);

    // 8) author output: identical rows = gnb_author @ W2_author + b2_author
    author_vec<<<1, 128, 0, stream>>>(gnb_a, W2_a, b2_a, avec);
    fill_rows<<<cdiv((long long)Na * 128, 256), 256, 0, stream>>>(avec, out_a, (long long)Na * 128);
}